// DPLRTransition_34385508171797
// MI455X (gfx1250) — compile-verified
//
#include <hip/hip_runtime.h>

typedef __attribute__((ext_vector_type(2))) float v2f;
typedef __attribute__((ext_vector_type(4))) float f32x4;
typedef __attribute__((ext_vector_type(8))) float v8f;

#define KD 128
#define VD 128
#define LDS_STRIDE 132   // 132 floats = 528 B row pitch: 16B-aligned, bank-skewed

__global__ __launch_bounds__(256)
void dplr_transition_kernel(const float* __restrict__ state,
                            const float* __restrict__ keys,
                            const float* __restrict__ values,
                            const float* __restrict__ gates,
                            const float* __restrict__ beta,
                            float* __restrict__ out)
{
    __shared__ float s_tile[KD * LDS_STRIDE];  // 67.6 KB staged S tile
    __shared__ float s_keys[KD];
    __shared__ float s_gate[KD];
    __shared__ float s_w[KD];                  // w = gates * keys
    __shared__ float s_vals[VD];
    __shared__ float s_kts[VD];                // ktS = w^T S
    __shared__ float s_beta;

    const int bh   = blockIdx.x;
    const int tid  = threadIdx.x;
    const size_t sbase   = (size_t)bh * (size_t)(KD * VD);
    const int    vecbase = bh * KD;

    // ---- stage 64KB state tile via CDNA5 async memory->LDS path ----
    // 4096 quads total; 256 threads x 16 quads, coalesced 16B per lane.
    #pragma unroll
    for (int i = 0; i < 16; ++i) {
        int q  = tid + i * 256;       // quad index 0..4095
        int k  = q >> 5;              // row
        int vq = (q & 31) << 2;       // starting column of quad
        unsigned long long ga =
            (unsigned long long)(const void*)(state + sbase + (size_t)q * 4);
        unsigned lds_b =
            (unsigned)(unsigned long long)(const void*)&s_tile[k * LDS_STRIDE + vq];
        asm volatile("global_load_async_to_lds_b128 %0, %1, off th:TH_LOAD_NT"
                     :: "v"(lds_b), "v"(ga) : "memory");
    }

    // ---- stage small vectors (regular path, overlaps with async DMA) ----
    if (tid < KD) {
        float kk = keys [vecbase + tid];
        float gg = gates[vecbase + tid];
        s_keys[tid] = kk;
        s_gate[tid] = gg;
        s_w[tid]    = kk * gg;
        s_vals[tid] = values[vecbase + tid];
    }
    if (tid == 0) s_beta = beta[bh];

    asm volatile("s_wait_asynccnt 0x0" ::: "memory");
    __syncthreads();

    // ---- ktS[v] = sum_k w[k] * S[k][v]  via V_WMMA_F32_16X16X4_F32 ----
    const int lane = tid & 31;
    const int wv   = tid >> 5;        // wave id 0..7 -> v-tile
    const int half = lane >> 4;       // K-pair select within chunk
    const int n    = lane & 15;       // column within v-tile
    const int n0   = wv << 4;

#if __has_builtin(__builtin_amdgcn_wmma_f32_16x16x4_f32)
    // A (16x4) = w chunk replicated across all 16 rows  -> every D row == ktS tile
    // B (4x16) = S[k0..k0+3][n0..n0+15] slab from LDS
    v8f acc = {};
    #pragma unroll 4
    for (int c = 0; c < 32; ++c) {
        int k0 = (c << 2) + (half << 1);
        v2f a, b;
        a.x = s_w[k0];
        a.y = s_w[k0 + 1];
        b.x = s_tile[(k0    ) * LDS_STRIDE + n0 + n];
        b.y = s_tile[(k0 + 1) * LDS_STRIDE + n0 + n];
        acc = __builtin_amdgcn_wmma_f32_16x16x4_f32(
                  /*neg_a=*/false, a, /*neg_b=*/false, b,
                  /*c_mod=*/(short)0, acc, /*reuse_a=*/false, /*reuse_b=*/false);
    }
    // all rows of D identical: acc[0] holds ktS[n0 + n] in every lane
    if (lane < 16) s_kts[n0 + n] = acc[0];
#else
    // VALU fallback: threads 0..127 each own one column
    if (tid < VD) {
        float acc = 0.f;
        #pragma unroll 8
        for (int k = 0; k < KD; ++k)
            acc += s_w[k] * s_tile[k * LDS_STRIDE + tid];
        s_kts[tid] = acc;
    }
#endif
    __syncthreads();

    // ---- out[k][v] = g[k]*S[k][v] + beta*keys[k]*(values[v] - ktS[v]) ----
    const float bta = s_beta;
    const int vq = (tid & 31) << 2;   // this thread's column quad
    const int kb = (tid >> 5) << 4;   // this thread's 16-row band
    f32x4 vv = *(const f32x4*)&s_vals[vq];
    f32x4 kt = *(const f32x4*)&s_kts[vq];
    f32x4 diff = vv - kt;

    f32x4* op = (f32x4*)(out + sbase);
    #pragma unroll
    for (int j = 0; j < 16; ++j) {
        int k = kb + j;
        float g  = s_gate[k];
        float bk = bta * s_keys[k];
        f32x4 s = *(const f32x4*)&s_tile[k * LDS_STRIDE + vq];
        f32x4 r = g * s + bk * diff;
        // write-once streaming output: bypass caches (TH_STORE_NT)
        __builtin_nontemporal_store(r, &op[(k << 5) + (vq >> 2)]);
    }
}

extern "C" void kernel_launch(void* const* d_in, const int* in_sizes, int n_in,
                              void* d_out, int out_size, void* d_ws, size_t ws_size,
                              hipStream_t stream) {
    const float* state  = (const float*)d_in[0];
    const float* keys   = (const float*)d_in[1];
    const float* values = (const float*)d_in[2];
    const float* gates  = (const float*)d_in[3];
    const float* beta   = (const float*)d_in[4];
    float* out = (float*)d_out;

    const int BH = in_sizes[4];  // beta has B*H elements -> number of (b,h) tiles
    dplr_transition_kernel<<<dim3(BH), dim3(256), 0, stream>>>(
        state, keys, values, gates, beta, out);
}